// TiedTropicalLowRankRecovery_69449621176345
// MI455X (gfx1250) — compile-verified
//
#include <hip/hip_runtime.h>
#include <math.h>

typedef __attribute__((ext_vector_type(2))) float v2f;
typedef __attribute__((ext_vector_type(8))) float v8f;

#define N_FEATURES 8192
#define MODEL_DIM  256
#define HEADS      4
#define CELLS      32
#define BATCH      4096

#define ROWS       8          // latent rows per block in reps_kernel
#define KC         32         // K-chunk staged in LDS for both GEMMs
#define BLK_N      64         // N tile per block (both GEMMs)
#define LDS_STRIDE 34         // padded: b64-aligned (even) + bank-conflict-free

// ---------------------------------------------------------------------------
// Stage 1: tropical routing -> reps.  8 latent rows per 128-thread block:
// one streamed W[t,d] value feeds 8 max-plus accumulators (cuts W re-read 8x).
// thread t -> router cell (h = t>>5, c = t&31).
// ---------------------------------------------------------------------------
__global__ __launch_bounds__(128)
void reps_kernel(const float* __restrict__ proj,
                 const float* __restrict__ rw,   // [H*C, D]
                 const float* __restrict__ rb,   // [H*C]
                 const float* __restrict__ cb,   // [H*C, D]
                 float* __restrict__ reps)       // [N_FEATURES, D]
{
    __shared__ float lat[ROWS][MODEL_DIM];       // 8 KB
    __shared__ float sc[ROWS][HEADS * CELLS];    // 4 KB
    __shared__ float s_gate[ROWS][HEADS];
    __shared__ int   s_w[ROWS][HEADS];
    __shared__ int   s_r[ROWS][HEADS];

    const int n0 = blockIdx.x * ROWS;
    const int t  = threadIdx.x;                  // 0..127

    #pragma unroll
    for (int p = 0; p < (ROWS * MODEL_DIM) / 128; ++p) {  // 16 iters
        const int flat = p * 128 + t;
        const int r = flat >> 8;
        const int d = flat & 255;
        lat[r][d] = proj[(size_t)(n0 + r) * MODEL_DIM + d];
    }
    __syncthreads();

    // max-plus tropical scores: stream this cell's W row once, apply to 8 rows
    const float* wrow = rw + (size_t)t * MODEL_DIM;
    float m[ROWS];
    #pragma unroll
    for (int r = 0; r < ROWS; ++r) m[r] = -3.402823466e38f;
    for (int d = 0; d < MODEL_DIM; ++d) {
        const float w = wrow[d];
        #pragma unroll
        for (int r = 0; r < ROWS; ++r)
            m[r] = fmaxf(m[r], lat[r][d] + w);
    }
    const float bb = rb[t];
    #pragma unroll
    for (int r = 0; r < ROWS; ++r) sc[r][t] = m[r] + bb;
    __syncthreads();

    // top-2 per (row, head); strict > keeps first occurrence (top_k tie rule)
    if (t < ROWS * HEADS) {                      // 32 threads
        const int r = t >> 2;
        const int h = t & 3;
        float best = -3.402823466e38f, second = -3.402823466e38f;
        int bi = 0, ri = 0;
        for (int c = 0; c < CELLS; ++c) {
            const float s = sc[r][h * CELLS + c];
            if (s > best)        { second = best; ri = bi; best = s; bi = c; }
            else if (s > second) { second = s; ri = c; }
        }
        const float margin = best - second;
        s_gate[r][h] = 1.0f / (1.0f + __expf(-margin));
        s_w[r][h] = bi;
        s_r[r][h] = ri;
    }
    __syncthreads();

    #pragma unroll
    for (int p = 0; p < (ROWS * MODEL_DIM) / 128; ++p) {
        const int flat = p * 128 + t;
        const int r = flat >> 8;
        const int d = flat & 255;
        float acc = lat[r][d];
        #pragma unroll
        for (int h = 0; h < HEADS; ++h) {
            const float g  = s_gate[r][h];
            const float wv = cb[(size_t)(h * CELLS + s_w[r][h]) * MODEL_DIM + d];
            const float rv = cb[(size_t)(h * CELLS + s_r[r][h]) * MODEL_DIM + d];
            acc += g * wv + (1.0f - g) * rv;
        }
        reps[(size_t)(n0 + r) * MODEL_DIM + d] = acc;
    }
}

// ---------------------------------------------------------------------------
// Stage 2: hidden[4096,256] = x[4096,8192] @ reps[8192,256]
// Block tile M=128 (8 waves x 16) x N=64, K chunked by 32.
// B chunk reps[k0:k0+32, n0:n0+64] staged into LDS *transposed* via
// global_load_async_to_lds_b32 (ASYNCcnt path, per-lane LDS scatter does the
// transpose), padded stride 34 -> aligned ds_load_b64 fragments, no bank
// conflicts. A fragments stream straight from global as contiguous v2f.
// ---------------------------------------------------------------------------
__global__ __launch_bounds__(256)
void gemm1_kernel(const float* __restrict__ x,
                  const float* __restrict__ reps,
                  float* __restrict__ hidden)
{
    __shared__ float bs[BLK_N * LDS_STRIDE];     // ~8.7 KB

    const int tid   = threadIdx.x;
    const int wave  = tid >> 5;
    const int lane  = tid & 31;
    const int m0    = blockIdx.y * 128 + wave * 16;
    const int n0    = blockIdx.x * BLK_N;
    const int r     = lane & 15;
    const int khalf = (lane >> 4) * 2;           // 0 or 2

    v8f acc[4] = {};
    const float* ap = x + (size_t)(m0 + r) * N_FEATURES + khalf;

    for (int k0 = 0; k0 < N_FEATURES; k0 += KC) {
        __syncthreads();                         // LDS reads of prev chunk done
        // async-stage B chunk: bs[nl*34 + kk] = reps[(k0+kk)*256 + n0+nl]
        #pragma unroll
        for (int p = 0; p < (KC * BLK_N) / 256; ++p) {   // 8 asyncs/thread
            const int flat = p * 256 + tid;
            const int nl = flat & (BLK_N - 1);   // contiguous global reads
            const int kk = flat >> 6;
            const unsigned ldsoff =
                (unsigned)(size_t)&bs[nl * LDS_STRIDE + kk];
            const float* gp = reps + (size_t)(k0 + kk) * MODEL_DIM + n0 + nl;
            asm volatile("global_load_async_to_lds_b32 %0, %1, off"
                         :: "v"(ldsoff), "v"(gp) : "memory");
        }
        // prefetch next A chunk while asyncs are in flight
        __builtin_prefetch(ap + k0 + KC, 0, 0);
        asm volatile("s_wait_asynccnt 0x0" ::: "memory");
        __syncthreads();                         // chunk visible block-wide

        #pragma unroll
        for (int kk = 0; kk < KC; kk += 4) {
            const v2f a = *(const v2f*)(ap + k0 + kk);
            #pragma unroll
            for (int j = 0; j < 4; ++j) {
                const v2f b =
                    *(const v2f*)&bs[(j * 16 + r) * LDS_STRIDE + kk + khalf];
                acc[j] = __builtin_amdgcn_wmma_f32_16x16x4_f32(
                    false, a, false, b, (short)0, acc[j], false, false);
            }
        }
    }

    // C layout: VGPR i -> M = i (+8 for lanes 16-31), N = lane&15
    const int cm = (lane >> 4) * 8;
    const int cn = lane & 15;
    #pragma unroll
    for (int j = 0; j < 4; ++j) {
        float* hp = hidden + (size_t)(m0 + cm) * MODEL_DIM + n0 + j * 16 + cn;
        #pragma unroll
        for (int i = 0; i < 8; ++i)
            hp[(size_t)i * MODEL_DIM] = acc[j][i];
    }
}

// ---------------------------------------------------------------------------
// Stage 3: out[4096,8192] = relu(hidden @ reps^T + bias)
// Same M=128 x N=64 block tile. B chunk reps[n0:n0+64, k0:k0+32] staged via
// coalesced loads + ds_store into the same padded-transpose layout.
// A fragments are contiguous v2f from hidden. Fused bias+ReLU epilogue.
// ---------------------------------------------------------------------------
__global__ __launch_bounds__(256)
void gemm2_kernel(const float* __restrict__ hidden,
                  const float* __restrict__ reps,
                  const float* __restrict__ bias,
                  float* __restrict__ out)
{
    __shared__ float bs[BLK_N * LDS_STRIDE];

    const int tid   = threadIdx.x;
    const int wave  = tid >> 5;
    const int lane  = tid & 31;
    const int m0    = blockIdx.y * 128 + wave * 16;
    const int n0    = blockIdx.x * BLK_N;
    const int r     = lane & 15;
    const int khalf = (lane >> 4) * 2;

    v8f acc[4] = {};
    const float* ap = hidden + (size_t)(m0 + r) * MODEL_DIM + khalf;

    for (int k0 = 0; k0 < MODEL_DIM; k0 += KC) {     // 8 chunks
        __syncthreads();
        // stage: bs[nl*34 + kk] = reps[(n0+nl)*256 + k0+kk]  (coalesced reads)
        #pragma unroll
        for (int p = 0; p < (KC * BLK_N) / 256; ++p) {
            const int flat = p * 256 + tid;
            const int kk = flat & (KC - 1);
            const int nl = flat >> 5;
            bs[nl * LDS_STRIDE + kk] =
                reps[(size_t)(n0 + nl) * MODEL_DIM + k0 + kk];
        }
        __syncthreads();

        #pragma unroll
        for (int kk = 0; kk < KC; kk += 4) {
            const v2f a = *(const v2f*)(ap + k0 + kk);
            #pragma unroll
            for (int j = 0; j < 4; ++j) {
                const v2f b =
                    *(const v2f*)&bs[(j * 16 + r) * LDS_STRIDE + kk + khalf];
                acc[j] = __builtin_amdgcn_wmma_f32_16x16x4_f32(
                    false, a, false, b, (short)0, acc[j], false, false);
            }
        }
    }

    const int cm = (lane >> 4) * 8;
    const int cn = lane & 15;
    #pragma unroll
    for (int j = 0; j < 4; ++j) {
        const int n = n0 + j * 16 + cn;
        const float bv = bias[n];
        float* op = out + (size_t)(m0 + cm) * N_FEATURES + n;
        #pragma unroll
        for (int i = 0; i < 8; ++i) {
            const float v = acc[j][i] + bv;
            op[(size_t)i * N_FEATURES] = fmaxf(v, 0.0f);
        }
    }
}

// ---------------------------------------------------------------------------
extern "C" void kernel_launch(void* const* d_in, const int* in_sizes, int n_in,
                              void* d_out, int out_size, void* d_ws, size_t ws_size,
                              hipStream_t stream)
{
    const float* x    = (const float*)d_in[0];  // [4096, 8192]
    const float* proj = (const float*)d_in[1];  // [8192, 256]
    const float* rw   = (const float*)d_in[2];  // [4, 32, 256]
    const float* rb   = (const float*)d_in[3];  // [4, 32]
    const float* cb   = (const float*)d_in[4];  // [4, 32, 256]
    const float* bias = (const float*)d_in[5];  // [8192]
    float* out = (float*)d_out;                 // [4096, 8192]

    float* reps   = (float*)d_ws;                                         // 8 MB
    float* hidden = (float*)((char*)d_ws +
                             (size_t)N_FEATURES * MODEL_DIM * sizeof(float)); // +4 MB

    // Stage 1: tropical routing -> reps (8 rows per block)
    reps_kernel<<<N_FEATURES / ROWS, 128, 0, stream>>>(proj, rw, rb, cb, reps);

    // Stage 2: hidden = x @ reps   (block tile 128x64, 8 waves)
    dim3 g1(MODEL_DIM / BLK_N, BATCH / 128);
    gemm1_kernel<<<g1, 256, 0, stream>>>(x, reps, hidden);

    // Stage 3: out = relu(hidden @ reps^T + bias)  (block tile 128x64)
    dim3 g2(N_FEATURES / BLK_N, BATCH / 128);
    gemm2_kernel<<<g2, 256, 0, stream>>>(hidden, reps, bias, out);
}